// PerceptronGlia_83846351553021
// MI455X (gfx1250) — compile-verified
//
#include <hip/hip_runtime.h>
#include <cstdint>

// ---------------------------------------------------------------------------
// PerceptronGlia pipeline for MI455X (gfx1250), wave32.
// One wave32 workgroup per batch row. Stage-1 h lives in registers
// (25 positions/lane), weight rows are streamed into LDS by the Tensor Data
// Mover (TDM), double-buffered, tracked with TENSORcnt. Final logits are
// produced with one V_WMMA_F32_16X16X4_F32 (diagonal-extraction identity).
// ---------------------------------------------------------------------------

#define BATCH 4096
#define N0    784
#define L0    263
#define W0ROW (782 * 3)   // 2346 floats per W0 step
#define L2N   121
#define W2ROW (254 * 3)   // 762 floats per W2 step
#define CH1   25          // stage-1 positions per lane (32*25 = 800 >= 784)
#define CH2   8           // stage-2 positions per lane (32*8 = 256)
#define NSTEP (L0 + L2N)  // 384 pipelined weight-row loads

// LDS layout in floats (single shared array => offsets are the LDS byte
// addresses the TDM needs, assuming static LDS starts at offset 0):
static constexpr int WBUF0 = 0;               // weight buffer 0 (2400 floats)
static constexpr int WBUF1 = 2400;            // weight buffer 1 (2400 floats)
static constexpr int HB0   = 4800;            // stage-2 h buffer 0 (260)
static constexpr int HB1   = 4800 + 260;      // stage-2 h buffer 1 (260)
static constexpr int SMEMF = 4800 + 520;      // ~21.3 KB total

typedef unsigned int       u32;
typedef unsigned long long u64;
typedef u32 u32x4 __attribute__((ext_vector_type(4)));
typedef int i32x8 __attribute__((ext_vector_type(8)));
typedef int i32x4 __attribute__((ext_vector_type(4)));

#if defined(__has_builtin)
#if __has_builtin(__builtin_amdgcn_tensor_load_to_lds) && __has_builtin(__builtin_amdgcn_s_wait_tensorcnt)
#define HAVE_TDM 1
#endif
#if __has_builtin(__builtin_amdgcn_wmma_f32_16x16x4_f32)
#define HAVE_WMMA 1
#endif
#endif
#ifndef HAVE_TDM
#define HAVE_TDM 0
#endif
#ifndef HAVE_WMMA
#define HAVE_WMMA 0
#endif

#if HAVE_TDM
#define WAIT_ROWS(n) __builtin_amdgcn_s_wait_tensorcnt(n)
#else
#define WAIT_ROWS(n) __syncthreads()
#endif

// Issue a 1-D TDM load of `n` f32 elements from `src` into LDS byte offset
// `lds_byte`. D# fields per CDNA5 ISA 8.3/8.4:
//   g0: count=1, lds_addr, global_addr[56:0], type=2
//   g1: data_size=2 (4B), tensor_dim0=n, tensor_dim1=1, tile_dim0=n,
//       tile_dim1=1, tensor_dim0_stride=n, everything else 0.
// This toolchain exposes the 6-arg builtin:
//   (u32x4 g0, i32x8 g1, i32x4 g2, i32x4 g3, i32x8 g4, i32 cpol)
__device__ __forceinline__ void load_row(float* lds_dst, u32 lds_byte,
                                         const float* src, u32 n, int lane) {
#if HAVE_TDM
  u64 ga = (u64)(uintptr_t)src;
  u32x4 g0;
  g0.x = 1u;                                         // count=1 (valid user D#)
  g0.y = lds_byte;                                   // LDS byte address
  g0.z = (u32)ga;                                    // global_addr[31:0]
  g0.w = (u32)((ga >> 32) & 0x01ffffffull) | (2u << 30); // addr[56:32] | type=2
  i32x8 g1;
  g1[0] = (int)(2u << 16);                           // data_size = 4 bytes
  g1[1] = (int)((n & 0xffffu) << 16);                // tensor_dim0[15:0]
  g1[2] = (int)((n >> 16) | (1u << 16));             // dim0[31:16] | tensor_dim1=1
  g1[3] = (int)((n & 0xffffu) << 16);                // tile_dim0 = n
  g1[4] = 1;                                         // tile_dim1 = 1
  g1[5] = (int)n;                                    // tensor_dim0_stride = n
  g1[6] = 0;
  g1[7] = 0;
  i32x4 z4 = {0, 0, 0, 0};                           // D# groups 2/3 disabled
  i32x8 z8 = {0, 0, 0, 0, 0, 0, 0, 0};
  __builtin_amdgcn_tensor_load_to_lds(g0, g1, z4, z4, z8, 0);
  (void)lds_dst; (void)lane;
#else
  for (u32 i = (u32)lane; i < n; i += 32u) lds_dst[i] = src[i];
#endif
}

__device__ __forceinline__ float elu(float v) {
  return (v > 0.f) ? v : expm1f(v);
}

__global__ __launch_bounds__(32) void glia_kernel(
    const float* __restrict__ x,  const float* __restrict__ W0,
    const float* __restrict__ Ws, const float* __restrict__ bs,
    const float* __restrict__ W2, const float* __restrict__ Wl,
    float* __restrict__ out) {
  __shared__ float smem[SMEMF];
  const int lane = (int)threadIdx.x;   // 0..31
  const int row  = (int)blockIdx.x;    // batch row

  // Zero the stage-1 weight tails [2346..2399] of both buffers once.
  // TDM only ever writes [0..2345] there, so no ordering issue.
  for (int i = lane; i < 2400 - W0ROW; i += 32) {
    smem[WBUF0 + W0ROW + i] = 0.f;
    smem[WBUF1 + W0ROW + i] = 0.f;
  }
  __syncthreads();

  // Pre-issue weight rows for steps 0 and 1 (double buffer prime).
  load_row(&smem[WBUF0], (u32)WBUF0 * 4u, W0 + (size_t)0 * W0ROW, W0ROW, lane);
  load_row(&smem[WBUF1], (u32)WBUF1 * 4u, W0 + (size_t)1 * W0ROW, W0ROW, lane);

  // ---- load x row into registers: lane owns j in [25*lane, 25*lane+24] ----
  float h[CH1];
#pragma unroll
  for (int i = 0; i < CH1; ++i) {
    int j = lane * CH1 + i;
    h[i] = (j < N0) ? x[(size_t)row * N0 + j] : 0.f;
  }

  // ---------------------- stage 1: 263 band steps --------------------------
  for (int l = 0; l < L0; ++l) {
    WAIT_ROWS(1);                       // loads complete in order => row l ready
    const float* wb = &smem[(l & 1) ? WBUF1 : WBUF0];
    // right halo: next lane's first two h values
    float e0 = __shfl_down(h[0], 1, 32);
    float e1 = __shfl_down(h[1], 1, 32);
    const int wbase = lane * (CH1 * 3);
#pragma unroll
    for (int i = 0; i < CH1; ++i) {     // ascending in-place update is safe
      float a = h[i];
      float b = (i < CH1 - 1) ? h[i + 1] : e0;
      float c = (i < CH1 - 2) ? h[i + 2] : ((i == CH1 - 2) ? e0 : e1);
      float v = a * wb[wbase + 3 * i] + b * wb[wbase + 3 * i + 1] +
                c * wb[wbase + 3 * i + 2];
      h[i] = elu(v);                    // masked (zero) weights => elu(0)=0 pad
    }
    int s = l + 2;                      // prefetch weight row for step l+2
    if (s < NSTEP) {
#if HAVE_TDM
      asm volatile("s_wait_dscnt 0" ::: "memory");   // drain LDS reads of this buf
#else
      __syncthreads();
#endif
      const float* src = (s < L0) ? (W0 + (size_t)s * W0ROW)
                                  : (W2 + (size_t)(s - L0) * W2ROW);
      u32 n    = (s < L0) ? (u32)W0ROW : (u32)W2ROW;
      u32 boff = (s & 1) ? (u32)WBUF1 : (u32)WBUF0;
      load_row(&smem[boff], boff * 4u, src, n, lane);
    }
  }

  // -------- middle layer: h[:256], pad 1 each side, Ws/bs, elu -------------
#pragma unroll
  for (int i = 0; i < CH1; ++i) {
    int j = lane * CH1 + i;
    if (j < 256) smem[HB0 + j] = h[i];
  }
  if (lane == 0) {                      // halo zeros for both stage-2 buffers
    smem[HB0 + 256] = 0.f; smem[HB0 + 257] = 0.f;
    smem[HB1 + 256] = 0.f; smem[HB1 + 257] = 0.f;
  }
  // zero stage-2 weight tails [762..767] in both buffers (in-flight W2 TDM
  // loads only write [0..761], disjoint)
  if (lane < 12) {
    int b = (lane < 6) ? WBUF0 : WBUF1;
    smem[b + W2ROW + (lane % 6)] = 0.f;
  }
  __syncthreads();
#pragma unroll
  for (int i = 0; i < CH2; ++i) {
    int j = lane * CH2 + i;
    float a = (j > 0) ? smem[HB0 + j - 1] : 0.f;
    float b = smem[HB0 + j];
    float c = smem[HB0 + j + 1];       // j=255 reads the zeroed halo
    float v = a * Ws[3 * j] + b * Ws[3 * j + 1] + c * Ws[3 * j + 2] + bs[j];
    smem[HB1 + j] = elu(v);
  }
  __syncthreads();

  // ---------------------- stage 2: 121 band steps --------------------------
  int cur = HB1, nxt = HB0;             // h double buffer in LDS
  for (int t = 0; t < L2N; ++t) {
    int s = L0 + t;
    if (t < L2N - 1) { WAIT_ROWS(1); } else { WAIT_ROWS(0); }
    const float* wb = &smem[(s & 1) ? WBUF1 : WBUF0];
#pragma unroll
    for (int i = 0; i < CH2; ++i) {
      int j = lane * CH2 + i;
      float v = smem[cur + j] * wb[3 * j] +
                smem[cur + j + 1] * wb[3 * j + 1] +
                smem[cur + j + 2] * wb[3 * j + 2];
      smem[nxt + j] = elu(v);
    }
    int s2 = s + 2;
    if (s2 < NSTEP) {
#if HAVE_TDM
      asm volatile("s_wait_dscnt 0" ::: "memory");
#else
      __syncthreads();
#endif
      u32 boff = (s2 & 1) ? (u32)WBUF1 : (u32)WBUF0;
      load_row(&smem[boff], boff * 4u, W2 + (size_t)(s2 - L0) * W2ROW,
               (u32)W2ROW, lane);
    }
    int tmp = cur; cur = nxt; nxt = tmp;
    __syncthreads();
  }

  // ---------------- final band over h[:12] + log_softmax -------------------
#if HAVE_WMMA
  {
    // Diagonal-extraction WMMA: D = A x B with
    //   A[m,k] = h[m+k]   (16x4 f32 A: lanes 0-15 give K=0,1; lanes 16-31 K=2,3)
    //   B[k,n] = Wl[n,k]  (4x16 f32 B: VGPR0 lanes0-15 k=0 / lanes16-31 k=1,
    //                                  VGPR1 lanes0-15 k=2 / lanes16-31 k=3)
    // => D[i,i] = sum_k h[i+k]*Wl[i,k] = logit[i] for i < 10.
    typedef float v2f __attribute__((ext_vector_type(2)));
    typedef float v8f __attribute__((ext_vector_type(8)));
    const int m = lane & 15;
    v2f a, b;
    if (lane < 16) { a.x = smem[cur + m];     a.y = smem[cur + m + 1]; }
    else           { a.x = smem[cur + m + 2]; a.y = 0.f; }
    float w0 = (m < 10) ? Wl[3 * m + 0] : 0.f;
    float w1 = (m < 10) ? Wl[3 * m + 1] : 0.f;
    float w2 = (m < 10) ? Wl[3 * m + 2] : 0.f;
    b.x = (lane < 16) ? w0 : w1;    // rows k=0 (lanes 0-15) / k=1 (16-31)
    b.y = (lane < 16) ? w2 : 0.f;   // rows k=2 (lanes 0-15) / k=3 (16-31)
    v8f c = {0.f, 0.f, 0.f, 0.f, 0.f, 0.f, 0.f, 0.f};
    c = __builtin_amdgcn_wmma_f32_16x16x4_f32(false, a, false, b,
                                              (short)0, c, false, false);
    // C/D layout: VGPR r, lane L -> M = r + 8*(L/16), N = L%16.
    // Diagonal i<8: lane i holds c[i]; i=8,9: lane i+16 holds c[i-8].
    if (lane < 8)                        smem[nxt + lane]        = c[lane];
    else if (lane >= 24 && lane < 26)    smem[nxt + (lane - 16)] = c[lane - 24];
    __syncthreads();
    if (lane == 0) {
      float lg[10];
      float mx = -3.402823466e38f;
#pragma unroll
      for (int i = 0; i < 10; ++i) { lg[i] = smem[nxt + i]; mx = fmaxf(mx, lg[i]); }
      float ssum = 0.f;
#pragma unroll
      for (int i = 0; i < 10; ++i) ssum += expf(lg[i] - mx);
      float lse = mx + logf(ssum);
#pragma unroll
      for (int i = 0; i < 10; ++i) out[(size_t)row * 10 + i] = lg[i] - lse;
    }
  }
#else
  if (lane == 0) {
    float hv[12];
#pragma unroll
    for (int k = 0; k < 12; ++k) hv[k] = smem[cur + k];
    float lg[10];
    float mx = -3.402823466e38f;
#pragma unroll
    for (int i = 0; i < 10; ++i) {
      lg[i] = hv[i] * Wl[3 * i] + hv[i + 1] * Wl[3 * i + 1] +
              hv[i + 2] * Wl[3 * i + 2];
      mx = fmaxf(mx, lg[i]);
    }
    float ssum = 0.f;
#pragma unroll
    for (int i = 0; i < 10; ++i) ssum += expf(lg[i] - mx);
    float lse = mx + logf(ssum);
#pragma unroll
    for (int i = 0; i < 10; ++i) out[(size_t)row * 10 + i] = lg[i] - lse;
  }
#endif
}

extern "C" void kernel_launch(void* const* d_in, const int* in_sizes, int n_in,
                              void* d_out, int out_size, void* d_ws,
                              size_t ws_size, hipStream_t stream) {
  (void)in_sizes; (void)n_in; (void)out_size; (void)d_ws; (void)ws_size;
  const float* x  = (const float*)d_in[0];
  const float* W0 = (const float*)d_in[1];
  const float* Ws = (const float*)d_in[2];
  const float* bs = (const float*)d_in[3];
  const float* W2 = (const float*)d_in[4];
  const float* Wl = (const float*)d_in[5];
  float* out = (float*)d_out;
  hipLaunchKernelGGL(glia_kernel, dim3(BATCH), dim3(32), 0, stream,
                     x, W0, Ws, bs, W2, Wl, out);
}